// DomainAwareStructuralPrior_90417651516042
// MI455X (gfx1250) — compile-verified
//
#include <hip/hip_runtime.h>
#include <stdint.h>

// Problem geometry (fixed by the reference: masks (128, 1024, 1024) fp32).
#define NIMG 128
#define H 1024
#define W 1024
#define ROWS_PER_WG 64
#define STRIPS (H / ROWS_PER_WG)      // 16 strips per image
#define TPB 256                        // 8 waves of 32
#define PPT 4                          // pixels (columns) per thread: 256*4 = 1024 = W
#define NR 8                           // LDS row ring (power of two) -> 32 KB
#define CHUNK 4                        // rows staged per pipeline step (double buffer in NR=8)

// Low 32 bits of a generic pointer into LDS == wave-relative LDS byte address
// (flat LDS addresses are {SHARED_BASE, offset[31:0]}; hardware truncates).
__device__ __forceinline__ unsigned lds_addr_u32(const void* p) {
  return (unsigned)(uintptr_t)p;
}

// CDNA5 async DMA: copy 16 bytes global -> LDS, tracked by ASYNCcnt. No VGPR data path.
__device__ __forceinline__ void async_copy_b128(unsigned lds_dst, const void* gsrc) {
  uint64_t ga = (uint64_t)(uintptr_t)gsrc;
  asm volatile("global_load_async_to_lds_b128 %0, %1, off"
               :: "v"(lds_dst), "v"(ga)
               : "memory");
}

__device__ __forceinline__ void wait_async_le4() {
  asm volatile("s_wait_asynccnt 4" ::: "memory");
}
__device__ __forceinline__ void wait_async_le0() {
  asm volatile("s_wait_asynccnt 0" ::: "memory");
}

__global__ __launch_bounds__(TPB) void structural_strip_kernel(
    const float* __restrict__ masks, float* __restrict__ partials) {
  __shared__ float rowbuf[NR][W];   // 32 KB row ring, async-DMA filled
  __shared__ float red[3][TPB];     // 3 KB reduction scratch

  const int wg    = blockIdx.x;
  const int img   = wg / STRIPS;
  const int strip = wg - img * STRIPS;
  const int r0    = strip * ROWS_PER_WG;
  const int t     = threadIdx.x;
  const int x0    = t * PPT;

  const float* base = masks + (size_t)img * H * W;

  const int gstart  = r0 - 1;                    // first staged row (halo)
  const int gend    = r0 + ROWS_PER_WG;          // last staged row (halo), inclusive
  const int nrows   = gend - gstart + 1;         // 66
  const int nchunks = (nrows + CHUNK - 1) / CHUNK;

  // Stage one chunk of rows into the LDS ring. Returns #async ops issued (per wave:
  // each in-range row is exactly one async b128 instruction per wave).
  auto issue_chunk = [&](int c) -> int {
    int issued = 0;
    const int cb = gstart + c * CHUNK;
    const int ce = min(cb + CHUNK - 1, gend);
    for (int g = cb; g <= ce; ++g) {
      const int slot = g & (NR - 1);             // -1 & 7 == 7, fine
      if (g >= 0 && g < H) {
        unsigned dst = lds_addr_u32(&rowbuf[slot][x0]);
        async_copy_b128(dst, base + (size_t)g * W + x0);
        ++issued;
      } else {
        // zero-pad halo row: store negatives so binarize() -> 0 (SAME zero padding)
        float4 neg = make_float4(-1.f, -1.f, -1.f, -1.f);
        *(float4*)&rowbuf[slot][x0] = neg;
      }
    }
    return issued;
  };

  float acc_area = 0.f, acc_conn = 0.f, acc_perim = 0.f;

  // Sliding 3-row window of per-column rowsum3 (rs) and center binaries (b),
  // kept in registers so each LDS row is read exactly once.
  float rs0[PPT] = {0.f, 0.f, 0.f, 0.f};   // row g-2
  float rs1[PPT] = {0.f, 0.f, 0.f, 0.f};   // row g-1
  float b1v[PPT] = {0.f, 0.f, 0.f, 0.f};   // binary of row g-1

  issue_chunk(0);
  int staged = 0;

  for (int c = 0; c < nchunks; ++c) {
    const int pend = (c + 1 < nchunks) ? issue_chunk(c + 1) : 0;
    // In-order per-wave async completion: outstanding <= pend  ==>  chunk c resident.
    if (pend == CHUNK) wait_async_le4();
    else               wait_async_le0();
    __syncthreads();

    const int cb = gstart + c * CHUNK;
    const int ce = min(cb + CHUNK - 1, gend);
    for (int g = cb; g <= ce; ++g) {
      const int slot = g & (NR - 1);
      const float* rowp = &rowbuf[slot][0];

      float4 v  = *(const float4*)(rowp + x0);
      float  vl = (t == 0)       ? -1.f : rowp[x0 - 1];
      float  vr = (t == TPB - 1) ? -1.f : rowp[x0 + PPT];

      const float c0 = v.x > 0.f ? 1.f : 0.f;
      const float c1 = v.y > 0.f ? 1.f : 0.f;
      const float c2 = v.z > 0.f ? 1.f : 0.f;
      const float c3 = v.w > 0.f ? 1.f : 0.f;
      const float cl = vl  > 0.f ? 1.f : 0.f;
      const float cr = vr  > 0.f ? 1.f : 0.f;

      float rs2[PPT], b2v[PPT];
      rs2[0] = cl + c0 + c1;  b2v[0] = c0;
      rs2[1] = c0 + c1 + c2;  b2v[1] = c1;
      rs2[2] = c1 + c2 + c3;  b2v[2] = c2;
      rs2[3] = c2 + c3 + cr;  b2v[3] = c3;

      if (staged >= 2) {   // emit stats for center row g-1 (always in [r0, r0+63])
#pragma unroll
        for (int i = 0; i < PPT; ++i) {
          const float b  = b1v[i];
          const float ns = rs0[i] + rs1[i] + rs2[i] - b;   // 8-neighbor count
          acc_area += b;
          acc_conn  = fmaf(ns, b, acc_conn);
          const float tt = fmaf(-4.f, b, ns);              // ns - 4*b
          acc_perim += fmaxf(tt, 0.f);
        }
      }
#pragma unroll
      for (int i = 0; i < PPT; ++i) { rs0[i] = rs1[i]; rs1[i] = rs2[i]; b1v[i] = b2v[i]; }
      ++staged;
    }
    __syncthreads();   // chunk c consumed; its ring slots may be DMA-overwritten next step
  }

  // Deterministic in-WG tree reduction (all values are exact small integers in fp32).
  red[0][t] = acc_area;  red[1][t] = acc_conn;  red[2][t] = acc_perim;
  __syncthreads();
  for (int s = TPB / 2; s > 0; s >>= 1) {
    if (t < s) {
      red[0][t] += red[0][t + s];
      red[1][t] += red[1][t + s];
      red[2][t] += red[2][t + s];
    }
    __syncthreads();
  }
  if (t == 0) {
    partials[(size_t)wg * 4 + 0] = red[0][0];
    partials[(size_t)wg * 4 + 1] = red[1][0];
    partials[(size_t)wg * 4 + 2] = red[2][0];
    partials[(size_t)wg * 4 + 3] = 0.f;
  }
}

__global__ void finalize_kernel(const float* __restrict__ partials,
                                float* __restrict__ out) {
  const int n = blockIdx.x * blockDim.x + threadIdx.x;
  if (n >= NIMG) return;

  float area = 0.f, conn = 0.f, perim = 0.f;
  for (int s = 0; s < STRIPS; ++s) {
    const size_t wg = (size_t)n * STRIPS + s;
    area  += partials[wg * 4 + 0];
    conn  += partials[wg * 4 + 1];
    perim += partials[wg * 4 + 2];
  }

  const float total_pixels = (float)(H * W);
  const float ar = area / total_pixels;
  const float area_score = (ar >= 0.001f && ar <= 0.5f) ? 1.f : 0.1f;

  const float safe_area = (area > 0.f) ? area : 1.f;
  const float comp_score = (conn > 0.f)
      ? fminf(1.f, 10.f / (conn / safe_area + 1e-6f))
      : 0.1f;

  const float par = perim / safe_area;
  const float shape_score = (area > 0.f)
      ? ((par <= 100.f) ? 1.f : fmaxf(0.1f, 100.f / (par + 1e-6f)))
      : 0.1f;

  const float v = 0.4f * area_score + 0.3f * comp_score + 0.3f * shape_score;
  out[n] = fmaxf(0.05f, v);
}

extern "C" void kernel_launch(void* const* d_in, const int* in_sizes, int n_in,
                              void* d_out, int out_size, void* d_ws, size_t ws_size,
                              hipStream_t stream) {
  (void)in_sizes; (void)n_in; (void)out_size; (void)ws_size;
  const float* masks   = (const float*)d_in[0];
  float*       out     = (float*)d_out;
  float*       partials = (float*)d_ws;   // NIMG*STRIPS*4 floats = 32 KB, fully rewritten each call

  structural_strip_kernel<<<dim3(NIMG * STRIPS), dim3(TPB), 0, stream>>>(masks, partials);
  finalize_kernel<<<dim3(1), dim3(128), 0, stream>>>(partials, out);
}